// ColorHistogramMatchingLoss_16312285790273
// MI455X (gfx1250) — compile-verified
//
#include <hip/hip_runtime.h>
#include <hip/hip_bf16.h>
#include <stdint.h>

// -------------------- problem geometry --------------------
#define NUM_BINS    64
#define HW_PIX      65536                 // 256*256 pixels per channel
#define NCH         24                    // 8 batch * 3 color channels
#define BLK_PER_CH  32                    // pixel-parallel blocks per channel
#define PIX_PER_BLK (HW_PIX / BLK_PER_CH) // 2048
#define TILE        2048                  // pixels staged in LDS per pass (8 KB)
#define THREADS     256                   // 8 wave32 per workgroup
#define CHUNKS_PER_WAVE (TILE / (4 * 8))  // 4 pixels/chunk, 8 waves -> 64

typedef __attribute__((ext_vector_type(2))) float v2f;
typedef __attribute__((ext_vector_type(8))) float v8f;
typedef int v4i_vs __attribute__((vector_size(16)));   // matches builtin param type

#if defined(__gfx1250__) && __has_builtin(__builtin_amdgcn_wmma_f32_16x16x4_f32)
#define HAVE_WMMA_F32X4 1
#endif

#if defined(__gfx1250__) && \
    __has_builtin(__builtin_amdgcn_global_load_async_to_lds_b128) && \
    __has_builtin(__builtin_amdgcn_s_wait_asynccnt)
#define HAVE_ASYNC_LDS 1
#endif

__device__ __forceinline__ float fast_exp2(float x) {
#if __has_builtin(__builtin_amdgcn_exp2f)
    return __builtin_amdgcn_exp2f(x);     // single v_exp_f32
#else
    return exp2f(x);
#endif
}

// Accumulate a 4-pixel x 16-bin weight tile into a 64-bin histogram group.
// WMMA path: D = ones(16x4) * B(4x16) + C  => every column of D is the
// 4-pixel sum for that bin; identical in all rows, so acc[0] holds it.
__device__ __forceinline__ void hist_accum(v8f& acc, float& s, v2f b, v2f ones) {
#ifdef HAVE_WMMA_F32X4
    (void)s;
    acc = __builtin_amdgcn_wmma_f32_16x16x4_f32(
        /*neg_a=*/false, ones, /*neg_b=*/false, b,
        /*c_mod=*/(short)0, acc, /*reuse_a=*/false, /*reuse_b=*/false);
#else
    (void)acc; (void)ones;
    s += b[0] + b[1];                      // per-lane-half partial
#endif
}

__device__ __forceinline__ float hist_extract(const v8f& acc, float s) {
#ifdef HAVE_WMMA_F32X4
    (void)s;
    return acc[0];                         // full 4-pixel-chunk sum per bin
#else
    return s + __shfl_xor(s, 16);          // combine lane halves
#endif
}

__global__ __launch_bounds__(THREADS) void zero_ws_kernel(float* __restrict__ p, int n) {
    int i = blockIdx.x * blockDim.x + threadIdx.x;
    if (i < n) p[i] = 0.0f;
}

__global__ __launch_bounds__(THREADS) void soft_hist_kernel(
        const float* __restrict__ pred,
        const float* __restrict__ tgt,
        float* __restrict__ hist /* [2][NCH][NUM_BINS], pre-zeroed */)
{
    __shared__ __align__(16) float tile[TILE];
    __shared__ float lhist[NUM_BINS];

    const int blk    = blockIdx.x;                 // 0 .. 2*NCH*BLK_PER_CH-1
    const int tensor = blk / (NCH * BLK_PER_CH);
    const int rem    = blk % (NCH * BLK_PER_CH);
    const int ch     = rem / BLK_PER_CH;
    const int seg    = rem % BLK_PER_CH;

    const float* __restrict__ src =
        (tensor == 0 ? pred : tgt) + (size_t)ch * HW_PIX + (size_t)seg * PIX_PER_BLK;
    float* __restrict__ hout = hist + (size_t)(tensor * NCH + ch) * NUM_BINS;

    const int t    = threadIdx.x;
    const int lane = t & 31;
    const int wave = t >> 5;

    if (t < NUM_BINS) lhist[t] = 0.0f;

    // Per-lane bin-center constants, one per 16-bin group.
    // exp(-(x-c)^2/(2*(1/64)^2)) = exp2(S*(x-c)^2), S = -2048*log2(e)
    // expanded: S*x^2 + k1*x + k0 with k1 = -2*S*c, k0 = S*c^2
    const int   n     = lane & 15;
    const float inv63 = 1.0f / 63.0f;                        // linspace(0,1,64)
    const float S     = -2048.0f * 1.4426950408889634f;
    const float c0 = (float)(n +  0) * inv63;
    const float c1 = (float)(n + 16) * inv63;
    const float c2 = (float)(n + 32) * inv63;
    const float c3 = (float)(n + 48) * inv63;
    const float k1_0 = -2.0f * S * c0, k0_0 = S * c0 * c0;
    const float k1_1 = -2.0f * S * c1, k0_1 = S * c1 * c1;
    const float k1_2 = -2.0f * S * c2, k0_2 = S * c2 * c2;
    const float k1_3 = -2.0f * S * c3, k0_3 = S * c3 * c3;

    v8f acc0 = {}, acc1 = {}, acc2 = {}, acc3 = {};
    float s0 = 0.0f, s1 = 0.0f, s2 = 0.0f, s3 = 0.0f;        // fallback partials
    const v2f ones = {1.0f, 1.0f};

    for (int tb = 0; tb < PIX_PER_BLK; tb += TILE) {
        __syncthreads();                                     // tile reuse fence
#ifdef HAVE_ASYNC_LDS
        {
            // two coalesced b128 async copies per thread: 256*2*16B = 8 KB
            const float* g = src + tb + t * 4;
            __builtin_amdgcn_global_load_async_to_lds_b128(
                (__attribute__((address_space(1))) v4i_vs*)(uintptr_t)g,
                (__attribute__((address_space(3))) v4i_vs*)(unsigned)(uintptr_t)&tile[t * 4],
                0, 0);
            __builtin_amdgcn_global_load_async_to_lds_b128(
                (__attribute__((address_space(1))) v4i_vs*)(uintptr_t)(g + 4 * THREADS),
                (__attribute__((address_space(3))) v4i_vs*)(unsigned)(uintptr_t)&tile[t * 4 + 4 * THREADS],
                0, 0);
            __builtin_amdgcn_s_wait_asynccnt(0);
        }
#else
        {
            const float4* g = (const float4*)(src + tb);
            float4*       l = (float4*)tile;
            l[t]           = g[t];
            l[t + THREADS] = g[t + THREADS];
        }
#endif
        __syncthreads();

        // each wave walks its private 256 pixels as 64 chunks of 4;
        // the ds_load_b128 below is wave-uniform => LDS broadcast read.
        const float4* chunks = (const float4*)tile + wave * CHUNKS_PER_WAVE;
        #pragma unroll 4
        for (int ck = 0; ck < CHUNKS_PER_WAVE; ++ck) {
            const float4 p = chunks[ck];
            // lanes 0-15 handle pixels {0,1} (K=0,1); lanes 16-31 pixels {2,3}
            const float x0 = (lane < 16) ? p.x : p.z;
            const float x1 = (lane < 16) ? p.y : p.w;
            // shared per-pixel term S*x^2 (amortized over all 64 bins)
            const float xx0 = x0 * x0 * S;
            const float xx1 = x1 * x1 * S;
            v2f b;

            b[0] = fast_exp2(fmaf(k1_0, x0, k0_0) + xx0);
            b[1] = fast_exp2(fmaf(k1_0, x1, k0_0) + xx1);
            hist_accum(acc0, s0, b, ones);

            b[0] = fast_exp2(fmaf(k1_1, x0, k0_1) + xx0);
            b[1] = fast_exp2(fmaf(k1_1, x1, k0_1) + xx1);
            hist_accum(acc1, s1, b, ones);

            b[0] = fast_exp2(fmaf(k1_2, x0, k0_2) + xx0);
            b[1] = fast_exp2(fmaf(k1_2, x1, k0_2) + xx1);
            hist_accum(acc2, s2, b, ones);

            b[0] = fast_exp2(fmaf(k1_3, x0, k0_3) + xx0);
            b[1] = fast_exp2(fmaf(k1_3, x1, k0_3) + xx1);
            hist_accum(acc3, s3, b, ones);
        }
    }

    // D-matrix layout: VGPR0 of acc holds bin (g*16 + (lane&15)) in every lane
    // (row M=0 for lanes 0-15, identical row M=8 for lanes 16-31).
    const float h0 = hist_extract(acc0, s0);
    const float h1 = hist_extract(acc1, s1);
    const float h2 = hist_extract(acc2, s2);
    const float h3 = hist_extract(acc3, s3);
    if (lane < 16) {
        atomicAdd(&lhist[n +  0], h0);
        atomicAdd(&lhist[n + 16], h1);
        atomicAdd(&lhist[n + 32], h2);
        atomicAdd(&lhist[n + 48], h3);
    }
    __syncthreads();
    if (t < NUM_BINS) atomicAdd(&hout[t], lhist[t]);
}

// normalize -> cumsum -> mean |pred_cum - target_cum| over 24*64 entries
__global__ __launch_bounds__(NUM_BINS) void finalize_kernel(
        const float* __restrict__ hist, float* __restrict__ out)
{
    __shared__ float red[NUM_BINS];
    const int t = threadIdx.x;        // bin index
    float total = 0.0f;
    for (int ch = 0; ch < NCH; ++ch) {
        const float* hp = hist + (size_t)ch * NUM_BINS;
        const float* ht = hist + (size_t)(NCH + ch) * NUM_BINS;
        float sp = 0.0f, st = 0.0f, cp = 0.0f, ct = 0.0f;
        for (int i = 0; i < NUM_BINS; ++i) { sp += hp[i]; st += ht[i]; }
        for (int i = 0; i <= t; ++i)       { cp += hp[i]; ct += ht[i]; }
        cp /= (sp + 1e-7f);
        ct /= (st + 1e-7f);
        total += fabsf(cp - ct);
    }
    red[t] = total;
    __syncthreads();
    for (int s = NUM_BINS / 2; s > 0; s >>= 1) {
        if (t < s) red[t] += red[t + s];
        __syncthreads();
    }
    if (t == 0) out[0] = red[0] / (float)(NCH * NUM_BINS);
}

extern "C" void kernel_launch(void* const* d_in, const int* in_sizes, int n_in,
                              void* d_out, int out_size, void* d_ws, size_t ws_size,
                              hipStream_t stream) {
    const float* pred = (const float*)d_in[0];   // (8,3,256,256) f32
    const float* tgt  = (const float*)d_in[1];   // (8,3,256,256) f32
    float* hist = (float*)d_ws;                  // [2][24][64] f32 = 12 KB
    const int nh = 2 * NCH * NUM_BINS;

    zero_ws_kernel<<<(nh + THREADS - 1) / THREADS, THREADS, 0, stream>>>(hist, nh);
    soft_hist_kernel<<<2 * NCH * BLK_PER_CH, THREADS, 0, stream>>>(pred, tgt, hist);
    finalize_kernel<<<1, NUM_BINS, 0, stream>>>(hist, (float*)d_out);
}